// GCN2BN1LIN2MLP3_61426622267904
// MI455X (gfx1250) — compile-verified
//
#include <hip/hip_runtime.h>
#include <hip/hip_bf16.h>
#include <math.h>

typedef __attribute__((ext_vector_type(2))) float v2f;
typedef __attribute__((ext_vector_type(8))) float v8f;

#define FDIM 128

// ---------------------------------------------------------------------------
// BatchNorm column statistics: block = 128 threads (one feature per thread),
// grid-stride over rows; coalesced row reads; one atomicAdd per thread/block.
// ---------------------------------------------------------------------------
__global__ void k_bnstats(const float* __restrict__ x, float* __restrict__ sum,
                          float* __restrict__ sqsum, int N) {
    const int f = threadIdx.x;  // 0..127
    float s = 0.f, q = 0.f;
    for (int row = blockIdx.x; row < N; row += gridDim.x) {
        float v = x[(size_t)row * FDIM + f];
        s += v;
        q += v * v;
    }
    atomicAdd(&sum[f], s);
    atomicAdd(&sqsum[f], q);
}

__global__ void k_bnfinal(const float* __restrict__ sum, const float* __restrict__ sqsum,
                          const float* __restrict__ gamma, const float* __restrict__ beta,
                          float* __restrict__ scale, float* __restrict__ shift, int N) {
    const int f = threadIdx.x;
    float invN = 1.0f / (float)N;
    float mean = sum[f] * invN;
    float var  = sqsum[f] * invN - mean * mean;   // biased variance
    float sc   = rsqrtf(var + 1e-5f) * gamma[f];
    scale[f] = sc;
    shift[f] = beta[f] - mean * sc;
}

// ---------------------------------------------------------------------------
// Degree / normalization: deg starts at 1 (self loop), +1 per incoming edge,
// then dinv = rsqrt(deg) in place.
// ---------------------------------------------------------------------------
__global__ void k_fill1(float* __restrict__ p, int n) {
    int i = blockIdx.x * blockDim.x + threadIdx.x;
    if (i < n) p[i] = 1.0f;
}

__global__ void k_degree(const int* __restrict__ dst, float* __restrict__ deg, int E) {
    int e = blockIdx.x * blockDim.x + threadIdx.x;
    if (e < E) atomicAdd(&deg[dst[e]], 1.0f);
}

__global__ void k_rsqrt_inplace(float* __restrict__ p, int n) {
    int i = blockIdx.x * blockDim.x + threadIdx.x;
    if (i < n) p[i] = rsqrtf(p[i]);
}

// ---------------------------------------------------------------------------
// WMMA fp32 GEMM:  C[M x NCOLS] = A[M x 128] * B[128 x NCOLS]
//   FUSE_BN : A element a -> a*scale[k] + shift[k]   (BN folded into conv1)
//   ADD_BIAS: epilogue += bias[col]
// One wave32 computes one 16x16 C tile via 32 x V_WMMA_F32_16X16X4_F32.
// A frag (16x4 f32): lane = grp*16+idx; v0=A[idx][k+2g], v1=A[idx][k+2g+1]
// B frag (4x16 f32): v0=B[k+2g][idx],  v1=B[k+2g+1][idx]
// C (16x16 f32): vgpr r holds C[r+8g][idx]
// Requires M % 16 == 0 (N=50000 = 3125*16). All branches wave-uniform or
// value-selects so EXEC is all-ones at every WMMA.
// ---------------------------------------------------------------------------
template <int NCOLS, bool FUSE_BN, bool ADD_BIAS>
__global__ __launch_bounds__(256) void k_gemm_wmma(
    const float* __restrict__ A, const float* __restrict__ B,
    const float* __restrict__ scale, const float* __restrict__ shift,
    const float* __restrict__ bias, float* __restrict__ C, int M) {
    const int wave = threadIdx.x >> 5;
    const int lane = threadIdx.x & 31;
    const int grp  = lane >> 4;   // 0|1
    const int idx  = lane & 15;

    const int Mtiles = M >> 4;
    const int mtile  = blockIdx.x * 8 + wave;   // uniform per wave
    if (mtile >= Mtiles) return;                // uniform branch, EXEC stays full
    const int ntile = blockIdx.y;

    const int row = (mtile << 4) + idx;         // A row read by this lane
    const int col = (ntile << 4) + idx;         // B/C column for this lane
    const float cmask = (col < NCOLS) ? 1.0f : 0.0f;
    const int colB = (col < NCOLS) ? col : (NCOLS - 1);  // clamped, always-valid load

    const float* __restrict__ Arow = A + (size_t)row * FDIM;

    v8f acc = {};
#pragma unroll
    for (int s = 0; s < FDIM / 4; ++s) {
        const int k = 4 * s + 2 * grp;
        float ax = Arow[k];
        float ay = Arow[k + 1];
        if (FUSE_BN) {
            ax = ax * scale[k]     + shift[k];
            ay = ay * scale[k + 1] + shift[k + 1];
        }
        v2f a; a.x = ax; a.y = ay;
        v2f b;
        b.x = B[(size_t)k       * NCOLS + colB] * cmask;
        b.y = B[(size_t)(k + 1) * NCOLS + colB] * cmask;
        acc = __builtin_amdgcn_wmma_f32_16x16x4_f32(
            /*neg_a=*/false, a, /*neg_b=*/false, b,
            /*c_mod=*/(short)0, acc, /*reuse_a=*/false, /*reuse_b=*/false);
    }

    float bv = 0.0f;
    if (ADD_BIAS && col < NCOLS) bv = bias[col];
    if (col < NCOLS) {
#pragma unroll
        for (int r = 0; r < 8; ++r) {
            const int m = (mtile << 4) + r + 8 * grp;
            C[(size_t)m * NCOLS + col] = acc[r] + bv;
        }
    }
}

// ---------------------------------------------------------------------------
// Edge scatter: acc[dst] += h[src] * dinv[src].  32 lanes per edge, float4
// per lane -> coalesced 512B read of the source row, 4 f32 atomics per lane.
// ---------------------------------------------------------------------------
__global__ void k_scatter(const float* __restrict__ h, const float* __restrict__ dinv,
                          const int* __restrict__ src, const int* __restrict__ dst,
                          float* __restrict__ acc, int E) {
    long long t = (long long)blockIdx.x * blockDim.x + threadIdx.x;
    long long total = (long long)E * 32;
    if (t >= total) return;
    const int e    = (int)(t >> 5);
    const int part = (int)(t & 31);
    const int s = src[e];
    const int d = dst[e];
    const float w = dinv[s];
    const float4 hv = *(const float4*)(h + (size_t)s * FDIM + part * 4);
    float* ap = acc + (size_t)d * FDIM + part * 4;
    atomicAdd(ap + 0, hv.x * w);
    atomicAdd(ap + 1, hv.y * w);
    atomicAdd(ap + 2, hv.z * w);
    atomicAdd(ap + 3, hv.w * w);
}

// out = relu( (acc + h*dinv[row]) * dinv[row] + bias[f] )
__global__ void k_conv_finalize(const float* __restrict__ acc, const float* __restrict__ h,
                                const float* __restrict__ dinv, const float* __restrict__ bias,
                                float* __restrict__ out, int N) {
    long long i = (long long)blockIdx.x * blockDim.x + threadIdx.x;
    long long total = (long long)N * FDIM;
    if (i >= total) return;
    const int row = (int)(i >> 7);
    const int f   = (int)(i & (FDIM - 1));
    const float di = dinv[row];
    float v = (acc[i] + h[i] * di) * di + bias[f];
    out[i] = fmaxf(v, 0.0f);
}

// ---------------------------------------------------------------------------
// log_softmax over 40 classes: one wave32 per row (lane covers c=lane and
// c=lane+32), wave shuffle reductions.
// ---------------------------------------------------------------------------
__global__ void k_logsoftmax40(float* __restrict__ z, int N) {
    const int wave = threadIdx.x >> 5;
    const int lane = threadIdx.x & 31;
    const int row = blockIdx.x * 8 + wave;
    if (row >= N) return;
    float* zr = z + (size_t)row * 40;
    const float v0 = zr[lane];                                 // lane < 32 < 40
    const float v1 = (lane < 8) ? zr[32 + lane] : -INFINITY;
    float m = fmaxf(v0, v1);
    for (int o = 16; o > 0; o >>= 1) m = fmaxf(m, __shfl_xor(m, o, 32));
    float s = __expf(v0 - m) + ((lane < 8) ? __expf(v1 - m) : 0.0f);
    for (int o = 16; o > 0; o >>= 1) s += __shfl_xor(s, o, 32);
    const float lg = m + __logf(s);
    zr[lane] = v0 - lg;
    if (lane < 8) zr[32 + lane] = v1 - lg;
}

// ---------------------------------------------------------------------------
extern "C" void kernel_launch(void* const* d_in, const int* in_sizes, int n_in,
                              void* d_out, int out_size, void* d_ws, size_t ws_size,
                              hipStream_t stream) {
    const float* x     = (const float*)d_in[0];
    const int*   ei    = (const int*)d_in[1];
    const float* gamma = (const float*)d_in[2];
    const float* beta  = (const float*)d_in[3];
    const float* W1  = (const float*)d_in[4];
    const float* b1  = (const float*)d_in[5];
    const float* W2  = (const float*)d_in[6];
    const float* b2  = (const float*)d_in[7];
    const float* Wm1 = (const float*)d_in[8];
    const float* bm1 = (const float*)d_in[9];
    const float* Wm2 = (const float*)d_in[10];
    const float* bm2 = (const float*)d_in[11];
    const float* Wm3 = (const float*)d_in[12];
    const float* bm3 = (const float*)d_in[13];

    const int N = in_sizes[0] / FDIM;   // 50000
    const int E = in_sizes[1] / 2;      // 800000
    const int*   src = ei;
    const int*   dst = ei + E;

    // Workspace carve-up (floats)
    float* ws    = (float*)d_ws;
    float* sum   = ws;           // 128
    float* sqsum = ws + 128;     // 128
    float* scale = ws + 256;     // 128
    float* shift = ws + 384;     // 128
    float* dinv  = ws + 512;     // N
    size_t off = 512 + (size_t)((N + 63) / 64) * 64;
    float* bufH   = ws + off; off += (size_t)N * FDIM;
    float* bufAcc = ws + off; off += (size_t)N * FDIM;
    float* bufR   = ws + off; off += (size_t)N * FDIM;

    const int Mtiles = N / 16;                      // 3125
    const dim3 gemmBlock(256);
    const dim3 gemmGrid128((Mtiles + 7) / 8, FDIM / 16);  // 391 x 8
    const dim3 gemmGrid40((Mtiles + 7) / 8, 3);           // 48 cols, guarded
    const size_t featBytes = (size_t)N * FDIM * sizeof(float);
    const long long scatWork = (long long)E * 32;
    const int scatGrid = (int)((scatWork + 255) / 256);
    const int featGrid = (int)(((long long)N * FDIM + 255) / 256);

    // --- BatchNorm folded into conv1 A-operand ---
    hipMemsetAsync(sum, 0, 256 * sizeof(float), stream);    // zero sum+sqsum
    k_bnstats<<<512, 128, 0, stream>>>(x, sum, sqsum, N);
    k_bnfinal<<<1, 128, 0, stream>>>(sum, sqsum, gamma, beta, scale, shift, N);

    // --- degree / dinv ---
    k_fill1<<<(N + 255) / 256, 256, 0, stream>>>(dinv, N);
    k_degree<<<(E + 255) / 256, 256, 0, stream>>>(dst, dinv, E);
    k_rsqrt_inplace<<<(N + 255) / 256, 256, 0, stream>>>(dinv, N);

    // --- conv1: h = BN(x) @ W1 ; aggregate ; relu(+b1) ---
    k_gemm_wmma<FDIM, true, false><<<gemmGrid128, gemmBlock, 0, stream>>>(
        x, W1, scale, shift, nullptr, bufH, N);
    hipMemsetAsync(bufAcc, 0, featBytes, stream);
    k_scatter<<<scatGrid, 256, 0, stream>>>(bufH, dinv, src, dst, bufAcc, E);
    k_conv_finalize<<<featGrid, 256, 0, stream>>>(bufAcc, bufH, dinv, b1, bufR, N);

    // --- conv2 ---
    k_gemm_wmma<FDIM, false, false><<<gemmGrid128, gemmBlock, 0, stream>>>(
        bufR, W2, nullptr, nullptr, nullptr, bufH, N);
    hipMemsetAsync(bufAcc, 0, featBytes, stream);
    k_scatter<<<scatGrid, 256, 0, stream>>>(bufH, dinv, src, dst, bufAcc, E);
    k_conv_finalize<<<featGrid, 256, 0, stream>>>(bufAcc, bufH, dinv, b2, bufR, N);

    // --- MLP: two 128x128 + one 128x40 (bias fused in epilogue) ---
    k_gemm_wmma<FDIM, false, true><<<gemmGrid128, gemmBlock, 0, stream>>>(
        bufR, Wm1, nullptr, nullptr, bm1, bufH, N);
    k_gemm_wmma<FDIM, false, true><<<gemmGrid128, gemmBlock, 0, stream>>>(
        bufH, Wm2, nullptr, nullptr, bm2, bufR, N);
    k_gemm_wmma<40, false, true><<<gemmGrid40, gemmBlock, 0, stream>>>(
        bufR, Wm3, nullptr, nullptr, bm3, (float*)d_out, N);

    // --- log_softmax over 40 classes ---
    k_logsoftmax40<<<(N + 7) / 8, 256, 0, stream>>>((float*)d_out, N);
}